// Converse2D_Base_46746424050083
// MI455X (gfx1250) — compile-verified
//
#include <hip/hip_runtime.h>

typedef float v2f __attribute__((ext_vector_type(2)));
typedef float v8f __attribute__((ext_vector_type(8)));

#define TWOPI 6.28318530717958647692f

// ---------------------------------------------------------------------------
// V_WMMA_F32_16X16X4_F32 wrapper: D(16x16) = A(16x4) x B(4x16) + C
// 8-arg pattern: (neg_a, A, neg_b, B, c_mod, C, reuse_a, reuse_b)
// ---------------------------------------------------------------------------
__device__ __forceinline__ v8f wmma4(v2f a, v2f b, v8f c) {
  return __builtin_amdgcn_wmma_f32_16x16x4_f32(false, a, false, b, (short)0, c,
                                               false, false);
}

// ---------------------------------------------------------------------------
// Async global->LDS staging (ASYNCcnt path); graceful fallback if the builtin
// is not declared by this toolchain. Builtin signature (probed via the
// round-1 diagnostic): (AS1 int*, AS3 int*, imm offset, imm cpol).
// ---------------------------------------------------------------------------
__device__ __forceinline__ void lds_async_load_f32(const float* g, float* l) {
#if __has_builtin(__builtin_amdgcn_global_load_async_to_lds_b32)
  __builtin_amdgcn_global_load_async_to_lds_b32(
      (__attribute__((address_space(1))) int*)g,
      (__attribute__((address_space(3))) int*)l, 0, 0);
#else
  *l = *g;
#endif
}

__device__ __forceinline__ void wait_async_lds() {
#if __has_builtin(__builtin_amdgcn_s_wait_asynccnt)
  __builtin_amdgcn_s_wait_asynccnt(0);
#else
  asm volatile("s_wait_asynccnt 0" ::: "memory");
#endif
  asm volatile("" ::: "memory");
}

// ---------------------------------------------------------------------------
// Wave-level DFT of length N = 16*R (R = 8 or 16), four-step via WMMA.
// Input:  z[n] stored in wave-private LDS tile at [n>>4][n&15], stride 17.
// Output: Rr/Ri in C/D layout; element (vgpr v, half h, lane l) is
//         spectral index k = (v + 8h) + 16*l   (for R=8 only l<8 is valid).
// dir = -1 forward, +1 inverse (unscaled).
// ---------------------------------------------------------------------------
template <int R>
__device__ __forceinline__ void wave_dft(float* zr, float* zi, float dir,
                                         v8f& Rr, v8f& Ri) {
  const int lane = threadIdx.x & 31;
  const int h = lane >> 4;
  const int l = lane & 15;
  const float wN = dir * (TWOPI / (16.0f * (float)R));  // full-size twiddle
  const float wR = dir * (TWOPI / (float)R);            // radix twiddle

  v8f Dr = {0.f, 0.f, 0.f, 0.f, 0.f, 0.f, 0.f, 0.f};
  v8f Di = {0.f, 0.f, 0.f, 0.f, 0.f, 0.f, 0.f, 0.f};

  // Stage 1: D[k0][n0] = sum_{n1<R} e^{dir*2pi*n1*k0/R} * z[n1][n0]
  // A = W1^T (16xR), rows k0 = lane l; B = Z (Rx16), K index = 4j + 2h + v.
#pragma unroll
  for (int j = 0; j < (R >> 2); ++j) {
    const int n1 = 4 * j + 2 * h;
    float s0, c0, s1, c1;
    __sincosf(wR * (float)(n1 * l), &s0, &c0);
    __sincosf(wR * (float)((n1 + 1) * l), &s1, &c1);
    v2f Ar = {c0, c1};
    v2f Ai = {s0, s1};
    v2f AiN = {-s0, -s1};
    v2f Br = {zr[n1 * 17 + l], zr[(n1 + 1) * 17 + l]};
    v2f Bi = {zi[n1 * 17 + l], zi[(n1 + 1) * 17 + l]};
    Dr = wmma4(Ar, Br, Dr);   // Re += Ar*Br
    Dr = wmma4(AiN, Bi, Dr);  // Re -= Ai*Bi
    Di = wmma4(Ar, Bi, Di);   // Im += Ar*Bi
    Di = wmma4(Ai, Br, Di);   // Im += Ai*Br
  }

  // Twiddle by e^{dir*2pi*n0*k0/N} and restage through LDS (16 rows now).
  asm volatile("" ::: "memory");
#pragma unroll
  for (int v = 0; v < 8; ++v) {
    const int k0 = v + 8 * h;
    float s, c;
    __sincosf(wN * (float)(k0 * l), &s, &c);
    const float tr = Dr[v] * c - Di[v] * s;
    const float ti = Dr[v] * s + Di[v] * c;
    zr[k0 * 17 + l] = tr;
    zi[k0 * 17 + l] = ti;
  }
  asm volatile("" ::: "memory");

  Rr = (v8f){0.f, 0.f, 0.f, 0.f, 0.f, 0.f, 0.f, 0.f};
  Ri = (v8f){0.f, 0.f, 0.f, 0.f, 0.f, 0.f, 0.f, 0.f};

  // Stage 2: R[k0][k1] = sum_{n0<16} D[k0][n0] * e^{dir*2pi*n0*k1/R}
  // A = D (16x16): row k0 = lane l, K = 4j + 2h + v (reads LDS columns).
  // B = W2 (16x16): row n0 = K index, col k1 = lane l.
#pragma unroll
  for (int j = 0; j < 4; ++j) {
    const int n0 = 4 * j + 2 * h;
    v2f Ar = {zr[l * 17 + n0], zr[l * 17 + n0 + 1]};
    v2f Ai = {zi[l * 17 + n0], zi[l * 17 + n0 + 1]};
    v2f AiN = {-Ai.x, -Ai.y};
    float s0, c0, s1, c1;
    __sincosf(wR * (float)(n0 * l), &s0, &c0);
    __sincosf(wR * (float)((n0 + 1) * l), &s1, &c1);
    v2f Br = {c0, c1};
    v2f Bi = {s0, s1};
    Rr = wmma4(Ar, Br, Rr);
    Rr = wmma4(AiN, Bi, Rr);
    Ri = wmma4(Ar, Bi, Ri);
    Ri = wmma4(Ai, Br, Ri);
  }
}

// ---------------------------------------------------------------------------
// Kernel 1: per-channel spectral mask  M[c][u][v]  (complex, 256x256)
//   FB at the 4 aliases computed directly from the 5x5 softmax PSF (25 sincos,
//   aliases via parity signs), then M = (conj(FB)*(1-Wt) + lam*K)/(lam*65536).
// ---------------------------------------------------------------------------
__global__ void k_precompute_M(const float* __restrict__ w,
                               const float* __restrict__ lam_p,
                               float* __restrict__ M) {
  const int c = blockIdx.y;
  const int idx = blockIdx.x * blockDim.x + threadIdx.x;  // 0..16383
  const int p = idx >> 7;
  const int q = idx & 127;

  __shared__ float sw[25];
  if (threadIdx.x < 25) sw[threadIdx.x] = w[c * 25 + threadIdx.x];
  __syncthreads();

  const float lam = lam_p[0];
  float Sr[2][2] = {{0.f, 0.f}, {0.f, 0.f}};
  float Si[2][2] = {{0.f, 0.f}, {0.f, 0.f}};
  const float step = -TWOPI / 256.0f;
#pragma unroll
  for (int k1 = 0; k1 < 5; ++k1) {
#pragma unroll
    for (int k2 = 0; k2 < 5; ++k2) {
      float s, cth;
      __sincosf(step * (float)(p * (k1 - 2) + q * (k2 - 2)), &s, &cth);
      const float wv = sw[k1 * 5 + k2];
      const float tr = wv * cth, ti = wv * s;
      const float s1 = (k1 & 1) ? -1.f : 1.f;  // alias sign (-1)^(a*k1)
      const float s2 = (k2 & 1) ? -1.f : 1.f;
      Sr[0][0] += tr;            Si[0][0] += ti;
      Sr[1][0] += s1 * tr;       Si[1][0] += s1 * ti;
      Sr[0][1] += s2 * tr;       Si[0][1] += s2 * ti;
      Sr[1][1] += s1 * s2 * tr;  Si[1][1] += s1 * s2 * ti;
    }
  }

  float su, cu, sv, cv;
  __sincosf(TWOPI * (float)p / 256.f, &su, &cu);
  __sincosf(TWOPI * (float)q / 256.f, &sv, &cv);
  // KU[a] = 1 + (-1)^a * e^{-2pi i p/256}
  const float KUr[2] = {1.f + cu, 1.f - cu}, KUi[2] = {-su, su};
  const float KVr[2] = {1.f + cv, 1.f - cv}, KVi[2] = {-sv, sv};

  float Kr[2][2], Ki[2][2];
  float invW = 0.f, Nr = 0.f, Ni = 0.f;
#pragma unroll
  for (int a = 0; a < 2; ++a) {
#pragma unroll
    for (int b = 0; b < 2; ++b) {
      Kr[a][b] = KUr[a] * KVr[b] - KUi[a] * KVi[b];
      Ki[a][b] = KUr[a] * KVi[b] + KUi[a] * KVr[b];
      const float m2 = Sr[a][b] * Sr[a][b] + Si[a][b] * Si[a][b];
      invW += m2;
      Nr += m2 + lam * (Sr[a][b] * Kr[a][b] - Si[a][b] * Ki[a][b]);
      Ni += lam * (Sr[a][b] * Ki[a][b] + Si[a][b] * Kr[a][b]);
    }
  }
  invW *= 0.25f; Nr *= 0.25f; Ni *= 0.25f;
  const float d = 1.f / (invW + lam);
  const float omWr = 1.f - Nr * d;
  const float omWi = -Ni * d;
  const float scale = 1.f / (lam * 65536.f);  // fold lambda div + ifft2 norm

#pragma unroll
  for (int a = 0; a < 2; ++a) {
#pragma unroll
    for (int b = 0; b < 2; ++b) {
      // conj(FB)*(1-Wt) + lam*K
      const float mr = Sr[a][b] * omWr + Si[a][b] * omWi + lam * Kr[a][b];
      const float mi = Sr[a][b] * omWi - Si[a][b] * omWr + lam * Ki[a][b];
      const size_t o =
          (((size_t)c * 256 + (p + 128 * a)) * 256 + (q + 128 * b)) * 2;
      M[o + 0] = mr * scale;
      M[o + 1] = mi * scale;
    }
  }
}

// ---------------------------------------------------------------------------
// Kernel 2: forward 128-pt DFT over rows of x (real input) -> Y (complex)
// one wave per (image,row); async global->LDS staging.
// ---------------------------------------------------------------------------
__global__ void k_fwd_rows(const float* __restrict__ x, float* __restrict__ Y) {
  __shared__ float sre[8][16 * 17];
  __shared__ float sim[8][16 * 17];
  const int warp = threadIdx.x >> 5, lane = threadIdx.x & 31;
  const int wid = blockIdx.x * 8 + warp;  // 0..32767
  const int img = wid >> 7, m = wid & 127;
  float* zr = &sre[warp][0];
  float* zi = &sim[warp][0];

  const float* row = x + ((size_t)img * 128 + m) * 128;
#pragma unroll
  for (int t = 0; t < 4; ++t) {
    const int n = lane + 32 * t;
    const int off = (n >> 4) * 17 + (n & 15);
    lds_async_load_f32(row + n, zr + off);
    zi[off] = 0.f;
  }
  wait_async_lds();

  v8f Rr, Ri;
  wave_dft<8>(zr, zi, -1.f, Rr, Ri);

  const int h = lane >> 4, l = lane & 15;
  if (l < 8) {
    float2* o = (float2*)Y + (size_t)img * 128 * 128 + (size_t)m * 128;
#pragma unroll
    for (int v = 0; v < 8; ++v) {
      const int k = v + 8 * h + 16 * l;
      o[k] = make_float2(Rr[v], Ri[v]);
    }
  }
}

// ---------------------------------------------------------------------------
// Kernel 3: forward 128-pt DFT over columns of Y -> Xf  (X = fft2_128(x))
// ---------------------------------------------------------------------------
__global__ void k_fwd_cols(const float* __restrict__ Y, float* __restrict__ Xf) {
  __shared__ float sre[8][16 * 17];
  __shared__ float sim[8][16 * 17];
  const int warp = threadIdx.x >> 5, lane = threadIdx.x & 31;
  const int wid = blockIdx.x * 8 + warp;
  const int img = wid >> 7, q = wid & 127;
  float* zr = &sre[warp][0];
  float* zi = &sim[warp][0];

  const float2* Yc = (const float2*)Y + (size_t)img * 128 * 128 + q;
#pragma unroll
  for (int t = 0; t < 4; ++t) {
    const int n = lane + 32 * t;
    const float2 v2 = Yc[(size_t)n * 128];
    const int off = (n >> 4) * 17 + (n & 15);
    zr[off] = v2.x;
    zi[off] = v2.y;
  }

  v8f Rr, Ri;
  wave_dft<8>(zr, zi, -1.f, Rr, Ri);

  const int h = lane >> 4, l = lane & 15;
  if (l < 8) {
    float2* o = (float2*)Xf + (size_t)img * 128 * 128 + q;
#pragma unroll
    for (int v = 0; v < 8; ++v) {
      const int k = v + 8 * h + 16 * l;
      o[(size_t)k * 128] = make_float2(Rr[v], Ri[v]);
    }
  }
}

// ---------------------------------------------------------------------------
// Kernel 4: inverse 256-pt DFT over v of FX[u][v] = M[c][u][v]*Xf[u&127][v&127]
// one wave per (image, u-row); results restaged via LDS for coalesced stores.
// ---------------------------------------------------------------------------
__global__ void k_inv_v(const float* __restrict__ M, const float* __restrict__ Xf,
                        float* __restrict__ T1) {
  __shared__ float sre[8][16 * 17];
  __shared__ float sim[8][16 * 17];
  const int warp = threadIdx.x >> 5, lane = threadIdx.x & 31;
  const int wid = blockIdx.x * 8 + warp;  // 0..65535
  const int img = wid >> 8, u = wid & 255;
  const int c = img & 63;
  float* zr = &sre[warp][0];
  float* zi = &sim[warp][0];

  const float2* Mrow = (const float2*)M + ((size_t)c * 256 + u) * 256;
  const float2* Xrow = (const float2*)Xf + ((size_t)img * 128 + (u & 127)) * 128;
#pragma unroll
  for (int t = 0; t < 8; ++t) {
    const int n = lane + 32 * t;
    const float2 mv = Mrow[n];
    const float2 xv = Xrow[n & 127];
    const int off = (n >> 4) * 17 + (n & 15);
    zr[off] = mv.x * xv.x - mv.y * xv.y;
    zi[off] = mv.x * xv.y + mv.y * xv.x;
  }

  v8f Rr, Ri;
  wave_dft<16>(zr, zi, 1.f, Rr, Ri);

  // restage in spectral order for coalesced global writes
  const int h = lane >> 4, l = lane & 15;
  asm volatile("" ::: "memory");
#pragma unroll
  for (int v = 0; v < 8; ++v) {
    zr[l * 17 + (v + 8 * h)] = Rr[v];  // row k>>4 == l, col k&15 == v+8h
    zi[l * 17 + (v + 8 * h)] = Ri[v];
  }
  asm volatile("" ::: "memory");

  float2* Trow = (float2*)T1 + ((size_t)img * 256 + u) * 256;
#pragma unroll
  for (int t = 0; t < 8; ++t) {
    const int k = lane + 32 * t;
    const int off = (k >> 4) * 17 + (k & 15);
    Trow[k] = make_float2(zr[off], zi[off]);
  }
}

// ---------------------------------------------------------------------------
// Kernel 5: inverse 256-pt DFT over u of T1[u][n]; keep real part, add bias.
// ---------------------------------------------------------------------------
__global__ void k_inv_u(const float* __restrict__ T1,
                        const float* __restrict__ bias,
                        float* __restrict__ out) {
  __shared__ float sre[8][16 * 17];
  __shared__ float sim[8][16 * 17];
  const int warp = threadIdx.x >> 5, lane = threadIdx.x & 31;
  const int wid = blockIdx.x * 8 + warp;  // 0..65535
  const int img = wid >> 8, n = wid & 255;
  const int c = img & 63;
  float* zr = &sre[warp][0];
  float* zi = &sim[warp][0];

  const float2* Tc = (const float2*)T1 + (size_t)img * 65536 + n;
#pragma unroll
  for (int t = 0; t < 8; ++t) {
    const int u = lane + 32 * t;
    const float2 v2 = Tc[(size_t)u * 256];
    const int off = (u >> 4) * 17 + (u & 15);
    zr[off] = v2.x;
    zi[off] = v2.y;
  }

  v8f Rr, Ri;
  wave_dft<16>(zr, zi, 1.f, Rr, Ri);

  const int h = lane >> 4, l = lane & 15;
  const float bv = bias[c];
  float* o = out + (size_t)img * 65536 + n;
#pragma unroll
  for (int v = 0; v < 8; ++v) {
    const int m = v + 8 * h + 16 * l;
    o[(size_t)m * 256] = Rr[v] + bv;
  }
}

// ---------------------------------------------------------------------------
extern "C" void kernel_launch(void* const* d_in, const int* in_sizes, int n_in,
                              void* d_out, int out_size, void* d_ws,
                              size_t ws_size, hipStream_t stream) {
  (void)in_sizes; (void)n_in; (void)out_size; (void)ws_size;
  const float* x    = (const float*)d_in[0];  // [4,64,128,128]
  const float* w    = (const float*)d_in[1];  // [1,64,5,5]
  const float* bias = (const float*)d_in[2];  // [1,64,1,1]
  const float* lam  = (const float*)d_in[3];  // [1,1,1,1]
  float* out = (float*)d_out;                 // [4,64,256,256]

  float* M  = (float*)d_ws;                          // 64*256*256*2 floats
  float* Xf = M + (size_t)64 * 256 * 256 * 2;        // 256*128*128*2 floats
  float* T1 = Xf + (size_t)256 * 128 * 128 * 2;      // 256*256*256*2 floats
  float* Y  = T1;  // forward-pass intermediate reuses T1 region (dead after K3)

  k_precompute_M<<<dim3(64, 64), 256, 0, stream>>>(w, lam, M);
  k_fwd_rows<<<4096, 256, 0, stream>>>(x, Y);
  k_fwd_cols<<<4096, 256, 0, stream>>>(Y, Xf);
  k_inv_v<<<8192, 256, 0, stream>>>(M, Xf, T1);
  k_inv_u<<<8192, 256, 0, stream>>>(T1, bias, out);
}